// GPRGNN_47107201303143
// MI455X (gfx1250) — compile-verified
//
#include <hip/hip_runtime.h>

// GPRGNN on MI455X (gfx1250, wave32).
//  - MLP via V_WMMA_F32_16X16X4_F32 (full fp32 precision, matches reference)
//  - 10-step PPR propagation: coalesced gather + global_atomic_add_f32 scatter,
//    everything (cur 25.6MB, idx 25.6MB, norm 12.8MB) L2-resident on 192MB L2.
//  - Self-loop contribution fused into finalize kernel (saves N*64 atomics/step).
// ws layout (floats): deg[N] | dinv[N] | dinv2[N] | norm[E] | bufA[N*64] | bufB[N*64]
// total ~65.2 MB.

#define N_NODES 100000
#define N_EDGES 3200000
#define DIN     512
#define DHID    256
#define DOUT    64
#define KSTEPS  10

typedef __attribute__((ext_vector_type(2))) float v2f;
typedef __attribute__((ext_vector_type(8))) float v8f;

// -------------------- prep kernels --------------------

// zero bufB (scatter target of step 0) and init deg with self-loop weight 1.0
__global__ __launch_bounds__(256) void k_init(float* __restrict__ deg,
                                              float* __restrict__ bufB) {
    int g = blockIdx.x * 256 + threadIdx.x;     // grid covers N_NODES*64
    bufB[g] = 0.0f;
    if (g < N_NODES) deg[g] = 1.0f;
}

__global__ __launch_bounds__(256) void k_count(const int* __restrict__ col,
                                               float* __restrict__ deg) {
    int e = blockIdx.x * 256 + threadIdx.x;
    if (e < N_EDGES) atomicAdd(&deg[col[e]], 1.0f);
}

__global__ __launch_bounds__(256) void k_dinv(const float* __restrict__ deg,
                                              float* __restrict__ dinv,
                                              float* __restrict__ dinv2) {
    int i = blockIdx.x * 256 + threadIdx.x;
    if (i < N_NODES) {
        float r = rsqrtf(deg[i]);   // deg >= 1 always (self-loop)
        dinv[i]  = r;
        dinv2[i] = r * r;
    }
}

__global__ __launch_bounds__(256) void k_norm(const int* __restrict__ row,
                                              const int* __restrict__ col,
                                              const float* __restrict__ dinv,
                                              float* __restrict__ norm) {
    int e = blockIdx.x * 256 + threadIdx.x;
    if (e < N_EDGES) norm[e] = dinv[row[e]] * dinv[col[e]];
}

// -------------------- fused MLP: h = relu(x@W1+b1)@W2+b2 --------------------
// One block = one 16-row slab. 16 waves each compute a 16x16 tile of the
// 16x256 hidden slab with f32 WMMA (K=512 in steps of 4), ReLU+bias into LDS.
// After barrier, waves 0..3 compute the 16x64 output tiles from LDS (K=256),
// writing bufA = h and hidden(d_out) = temp[0]*h.
#define LDS_STRIDE 260   // 16 rows x 260 floats; stride%64==4 -> conflict-free col reads

__global__ __launch_bounds__(512)
void k_mlp(const float* __restrict__ x,  const float* __restrict__ W1,
           const float* __restrict__ b1, const float* __restrict__ W2,
           const float* __restrict__ b2, const float* __restrict__ temp,
           float* __restrict__ bufA,     float* __restrict__ hidden) {
    __shared__ float lds[16 * LDS_STRIDE];

    const int tid  = threadIdx.x;
    const int wave = tid >> 5;          // 0..15
    const int lane = tid & 31;
    const int l15  = lane & 15;
    const int kh   = lane >> 4;         // 0|1: which K/M half this lane covers
    const int m0   = blockIdx.x * 16;   // 100000 % 16 == 0 -> no tail

    // ---- phase 1: h1 tile, column block `wave` (cols wave*16 .. wave*16+15)
    v8f acc = {0.f, 0.f, 0.f, 0.f, 0.f, 0.f, 0.f, 0.f};
    const float* xrow  = x  + (size_t)(m0 + l15) * DIN;   // A: M = l15
    const float* w1col = W1 + wave * 16 + l15;            // B: N = l15
    for (int k0 = 0; k0 < DIN; k0 += 4) {
        int ka = k0 + kh * 2;                             // lane's K pair
        v2f a = *(const v2f*)(xrow + ka);                 // A[l15][ka..ka+1]
        v2f b;
        b.x = w1col[(size_t)ka * DHID];                   // B[ka][l15]
        b.y = w1col[(size_t)(ka + 1) * DHID];             // B[ka+1][l15]
        acc = __builtin_amdgcn_wmma_f32_16x16x4_f32(
                  false, a, false, b, (short)0, acc, false, false);
    }
    {
        float bias = b1[wave * 16 + l15];
        int rbase = kh * 8;   // C layout: vgpr r -> row r (lanes 0-15), r+8 (16-31)
        #pragma unroll
        for (int r = 0; r < 8; ++r) {
            float v = acc[r] + bias;
            v = v > 0.0f ? v : 0.0f;                      // ReLU
            lds[(rbase + r) * LDS_STRIDE + wave * 16 + l15] = v;
        }
    }
    __syncthreads();

    // ---- phase 2: output tiles (64 cols = 4 tiles), waves 0..3
    if (wave < 4) {
        v8f acc2 = {0.f, 0.f, 0.f, 0.f, 0.f, 0.f, 0.f, 0.f};
        const float* w2col = W2 + wave * 16 + l15;
        for (int k0 = 0; k0 < DHID; k0 += 4) {
            int ka = k0 + kh * 2;
            v2f a = *(const v2f*)(&lds[l15 * LDS_STRIDE + ka]);  // A = h1 slab
            v2f b;
            b.x = w2col[ka * DOUT];
            b.y = w2col[(ka + 1) * DOUT];
            acc2 = __builtin_amdgcn_wmma_f32_16x16x4_f32(
                       false, a, false, b, (short)0, acc2, false, false);
        }
        float bias = b2[wave * 16 + l15];
        float t0   = temp[0];                              // uniform -> s_load
        int rbase = kh * 8;
        #pragma unroll
        for (int r = 0; r < 8; ++r) {
            float v = acc2[r] + bias;
            int idx = (m0 + rbase + r) * DOUT + wave * 16 + l15;
            bufA[idx]   = v;        // cur = h
            hidden[idx] = t0 * v;   // hidden = temp[0]*h (fresh every call)
        }
    }
}

// -------------------- propagation --------------------
// 64 threads per edge -> fully coalesced gather + atomic scatter.
// Edge id is wave-uniform (wave covers 32 consecutive g, 64 | slice alignment),
// so force it scalar: row/col/norm become s_loads.
__global__ __launch_bounds__(256)
void k_scatter(const int* __restrict__ row, const int* __restrict__ col,
               const float* __restrict__ norm, const float* __restrict__ src,
               float* __restrict__ dst) {
    int g = blockIdx.x * 256 + threadIdx.x;      // grid == N_EDGES*64 exactly
    int e = __builtin_amdgcn_readfirstlane(g >> 6);
    int c = g & 63;
    int r  = row[e];
    int cl = col[e];
    float v = norm[e] * src[r * DOUT + c];
    atomicAdd(&dst[cl * DOUT + c], v);
}

// dst += self-loop term; hidden += gamma*dst; zero src (becomes next dst).
__global__ __launch_bounds__(256)
void k_finalize(float* __restrict__ dst, float* __restrict__ src,
                const float* __restrict__ dinv2, float* __restrict__ hidden,
                const float* __restrict__ temp, int kidx) {
    int g = blockIdx.x * 256 + threadIdx.x;      // grid == N_NODES*64 exactly
    int i = __builtin_amdgcn_readfirstlane(g >> 6);
    float v = dst[g] + dinv2[i] * src[g];
    dst[g] = v;
    hidden[g] += temp[kidx] * v;                 // uniform s_load of gamma
    src[g] = 0.0f;
}

// -------------------- launch --------------------
extern "C" void kernel_launch(void* const* d_in, const int* in_sizes, int n_in,
                              void* d_out, int out_size, void* d_ws, size_t ws_size,
                              hipStream_t stream) {
    const float* x    = (const float*)d_in[0];
    const int*   ei   = (const int*)  d_in[1];   // [2, E] int32
    const float* W1   = (const float*)d_in[2];
    const float* b1   = (const float*)d_in[3];
    const float* W2   = (const float*)d_in[4];
    const float* b2   = (const float*)d_in[5];
    const float* temp = (const float*)d_in[6];   // 11 gammas
    const int* row = ei;                         // sources  (gather)
    const int* col = ei + N_EDGES;               // targets  (scatter)
    float* hidden = (float*)d_out;

    float* deg   = (float*)d_ws;                 // N
    float* dinv  = deg   + N_NODES;              // N
    float* dinv2 = dinv  + N_NODES;              // N
    float* normw = dinv2 + N_NODES;              // E
    float* bufA  = normw + N_EDGES;              // N*64
    float* bufB  = bufA  + (size_t)N_NODES * DOUT;

    k_init <<<(N_NODES * DOUT) / 256, 256, 0, stream>>>(deg, bufB);
    k_count<<<(N_EDGES + 255) / 256, 256, 0, stream>>>(col, deg);
    k_dinv <<<(N_NODES + 255) / 256, 256, 0, stream>>>(deg, dinv, dinv2);
    k_norm <<<(N_EDGES + 255) / 256, 256, 0, stream>>>(row, col, dinv, normw);

    k_mlp<<<N_NODES / 16, 512, 0, stream>>>(x, W1, b1, W2, b2, temp, bufA, hidden);

    for (int k = 0; k < KSTEPS; ++k) {
        float* src = (k & 1) ? bufB : bufA;
        float* dst = (k & 1) ? bufA : bufB;
        k_scatter <<<(N_EDGES * 64) / 256, 256, 0, stream>>>(row, col, normw, src, dst);
        k_finalize<<<(N_NODES * DOUT) / 256, 256, 0, stream>>>(dst, src, dinv2,
                                                               hidden, temp, k + 1);
    }
}